// gnn_binary_classifier_62672162784007
// MI455X (gfx1250) — compile-verified
//
#include <hip/hip_runtime.h>
#include <math.h>

typedef float v2f __attribute__((ext_vector_type(2)));
typedef float v8f __attribute__((ext_vector_type(8)));

#define S        4096            // STATE_DIM
#define NNODES   4
#define TWO_S    (2 * S)         // 8192
#define GATES3   (3 * S)         // 12288
#define ENC_DIM  (NNODES * S)    // 16384
#define HID      4096            // HIDDEN_DIM
#define NW       8               // waves per workgroup in the WMMA GEMV
#define KUNROLL  32              // K elements per outer iteration (8 WMMAs)

// ---------------------------------------------------------------------------
// a[n, k] = (k < S ? A_out : A_in)[n, :] . h[:, k % S]      a: [4, 8192]
// ---------------------------------------------------------------------------
__global__ void propagate_kernel(const float* __restrict__ A_out,
                                 const float* __restrict__ A_in,
                                 const float* __restrict__ h,
                                 float* __restrict__ a) {
  int k = blockIdx.x * blockDim.x + threadIdx.x;
  if (k >= TWO_S) return;
  const float* __restrict__ A = (k < S) ? A_out : A_in;
  int kk = k & (S - 1);
  float h0 = h[0 * S + kk], h1 = h[1 * S + kk];
  float h2 = h[2 * S + kk], h3 = h[3 * S + kk];
#pragma unroll
  for (int n = 0; n < NNODES; ++n) {
    a[(size_t)n * TWO_S + k] =
        A[n * 4 + 0] * h0 + A[n * 4 + 1] * h1 + A[n * 4 + 2] * h2 + A[n * 4 + 3] * h3;
  }
}

// ---------------------------------------------------------------------------
// D[m, n] = sum_k X[m, k] * W[n, k]  (+ bias[n]),   m < M <= 16, row-major.
// One workgroup per 16-wide N tile; NW waves split K; f32 WMMA 16x16x4.
//
// Fragment layouts (CDNA5 ISA 7.12.2, 32-bit A 16x4 / B 4x16):
//   lane L holds K = k0 + 2*(L>>4) + {0,1} of row (L&15)  -> one b64 per lane.
// C/D 16x16 f32: VGPR v, lanes 0-15 -> M=v, lanes 16-31 -> M=v+8.
// The K-loop trip count is uniform (scalar) so EXEC stays all-ones around
// the WMMAs and the loop uses s_cmp/s_cbranch instead of exec masking.
// ---------------------------------------------------------------------------
__global__ void __launch_bounds__(NW * 32)
wmma_gemv_kernel(const float* __restrict__ X,   // [M, K] (rows >= M never stored)
                 const float* __restrict__ W,   // [N, K] row-major
                 const float* __restrict__ bias,// [N] or nullptr
                 float* __restrict__ D,         // [M, N] row-major
                 int M, int N, int K) {
  const int lane = threadIdx.x & 31;
  const int wave = threadIdx.x >> 5;
  const int n0   = blockIdx.x * 16;
  const int r    = lane & 15;           // tile row (A) / W row within tile (B)
  const int koff = 2 * (lane >> 4);     // 0 or 2

  const int kPerWave = K / NW;          // all K here divisible by NW*KUNROLL
  const int iters    = kPerWave / KUNROLL;  // uniform scalar trip count
  const int kBeg     = wave * kPerWave;

  const int xRowIdx = (r < M) ? r : 0;  // safe row for padding lanes
  const float* __restrict__ wp = W + (size_t)(n0 + r) * K + koff + kBeg;
  const float* __restrict__ xp = X + (size_t)xRowIdx * K + koff + kBeg;

  v8f acc = {0.f, 0.f, 0.f, 0.f, 0.f, 0.f, 0.f, 0.f};

  for (int it = 0; it < iters; ++it) {
    __builtin_prefetch(wp + 2048, 0, 0);  // ~8KB ahead of the W stream
#pragma unroll
    for (int u = 0; u < KUNROLL; u += 4) {
      v2f bfrag = *(const v2f*)(wp + u);
      v2f afrag = *(const v2f*)(xp + u);
      // 8 args: (neg_a, A, neg_b, B, c_mod, C, reuse_a, reuse_b)
      acc = __builtin_amdgcn_wmma_f32_16x16x4_f32(false, afrag, false, bfrag,
                                                  (short)0, acc, false, false);
    }
    wp += KUNROLL;
    xp += KUNROLL;
  }

  // cross-wave K reduction through LDS
  __shared__ v8f red[NW][32];
  red[wave][lane] = acc;
  __syncthreads();
  if (wave == 0) {
    v8f s = red[0][lane];
#pragma unroll
    for (int w = 1; w < NW; ++w) s += red[w][lane];
    const int col  = lane & 15;
    const int half = lane >> 4;
    const float b  = bias ? bias[n0 + col] : 0.f;
#pragma unroll
    for (int v = 0; v < 8; ++v) {
      int m = v + 8 * half;
      if (m < M) D[(size_t)m * N + n0 + col] = s[v] + b;
    }
  }
}

// ---------------------------------------------------------------------------
// GRU gate math: r,z,n gates, h_out = (1-z)*n + z*h
// ---------------------------------------------------------------------------
__global__ void gru_gate_kernel(const float* __restrict__ gi,  // [4, 12288]
                                const float* __restrict__ gh,  // [4, 12288]
                                const float* __restrict__ h,   // [4, 4096]
                                float* __restrict__ h_out) {   // [4, 4096]
  int idx = blockIdx.x * blockDim.x + threadIdx.x;
  if (idx >= NNODES * S) return;
  int n = idx / S;
  int c = idx & (S - 1);
  const float* __restrict__ gin = gi + (size_t)n * GATES3;
  const float* __restrict__ ghn = gh + (size_t)n * GATES3;
  float ir = gin[c], iz = gin[S + c], inn = gin[2 * S + c];
  float hr = ghn[c], hz = ghn[S + c], hnn = ghn[2 * S + c];
  float rg = 1.f / (1.f + expf(-(ir + hr)));
  float zg = 1.f / (1.f + expf(-(iz + hz)));
  float ng = tanhf(inn + rg * hnn);
  h_out[idx] = (1.f - zg) * ng + zg * h[idx];
}

// ---------------------------------------------------------------------------
// fc2 (2 x 4096) + log_softmax, single block
// ---------------------------------------------------------------------------
__global__ void head_kernel(const float* __restrict__ x,      // [4096] (fc1 out)
                            const float* __restrict__ fc2_w,  // [2, 4096]
                            const float* __restrict__ fc2_b,  // [2]
                            float* __restrict__ out) {        // [2]
  __shared__ float s0[256], s1[256];
  int tid = threadIdx.x;
  float a0 = 0.f, a1 = 0.f;
  for (int j = tid; j < HID; j += 256) {
    float v = x[j];
    a0 += v * fc2_w[j];
    a1 += v * fc2_w[HID + j];
  }
  s0[tid] = a0; s1[tid] = a1;
  __syncthreads();
  for (int off = 128; off > 0; off >>= 1) {
    if (tid < off) { s0[tid] += s0[tid + off]; s1[tid] += s1[tid + off]; }
    __syncthreads();
  }
  if (tid == 0) {
    float x0 = s0[0] + fc2_b[0];
    float x1 = s1[0] + fc2_b[1];
    float mx = fmaxf(x0, x1);
    float lse = mx + logf(expf(x0 - mx) + expf(x1 - mx));
    out[0] = x0 - lse;
    out[1] = x1 - lse;
  }
}

// ---------------------------------------------------------------------------
extern "C" void kernel_launch(void* const* d_in, const int* in_sizes, int n_in,
                              void* d_out, int out_size, void* d_ws, size_t ws_size,
                              hipStream_t stream) {
  (void)in_sizes; (void)n_in; (void)out_size; (void)ws_size;

  const float* A_out = (const float*)d_in[0];
  const float* A_in  = (const float*)d_in[1];
  const float* annot = (const float*)d_in[2];   // [4, 4096]
  const float* W_ih  = (const float*)d_in[3];   // [12288, 8192]
  const float* W_hh  = (const float*)d_in[4];   // [12288, 4096]
  const float* fc1_w = (const float*)d_in[5];   // [4096, 16384]
  const float* fc1_b = (const float*)d_in[6];   // [4096]
  const float* fc2_w = (const float*)d_in[7];   // [2, 4096]
  const float* fc2_b = (const float*)d_in[8];   // [2]
  float* out = (float*)d_out;

  // workspace carve-up (all float32): ~672 KB total
  float* ws     = (float*)d_ws;
  float* a_buf  = ws;                                // 4 * 8192
  float* gi_buf = a_buf  + (size_t)NNODES * TWO_S;   // 4 * 12288
  float* gh_buf = gi_buf + (size_t)NNODES * GATES3;  // 4 * 12288
  float* hA     = gh_buf + (size_t)NNODES * GATES3;  // 4 * 4096
  float* hB     = hA     + (size_t)NNODES * S;       // 4 * 4096
  float* fc1o   = hB     + (size_t)NNODES * S;       // 4096

  const float* h_cur = annot;
  float* bufs[2] = {hA, hB};

  for (int step = 0; step < 8; ++step) {
    float* h_next = bufs[step & 1];
    propagate_kernel<<<TWO_S / 256, 256, 0, stream>>>(A_out, A_in, h_cur, a_buf);
    // gi = a @ W_ih^T   (K = 8192, K/NW = 1024, 32 outer iters)
    wmma_gemv_kernel<<<GATES3 / 16, NW * 32, 0, stream>>>(
        a_buf, W_ih, nullptr, gi_buf, NNODES, GATES3, TWO_S);
    // gh = h @ W_hh^T   (K = 4096, K/NW = 512, 16 outer iters)
    wmma_gemv_kernel<<<GATES3 / 16, NW * 32, 0, stream>>>(
        h_cur, W_hh, nullptr, gh_buf, NNODES, GATES3, S);
    gru_gate_kernel<<<(NNODES * S) / 256, 256, 0, stream>>>(gi_buf, gh_buf, h_cur, h_next);
    h_cur = h_next;
  }

  // enc = h.reshape(1, 16384); fc1: [1,16384] @ [16384,4096]^T + b
  wmma_gemv_kernel<<<HID / 16, NW * 32, 0, stream>>>(
      h_cur, fc1_w, fc1_b, fc1o, 1, HID, ENC_DIM);

  head_kernel<<<1, 256, 0, stream>>>(fc1o, fc2_w, fc2_b, out);
}